// PGAgent_to_87668872446277
// MI455X (gfx1250) — compile-verified
//
#include <hip/hip_runtime.h>
#include <hip/hip_bf16.h>

// Problem constants (from reference)
#define B_   8192
#define NY_  8
#define MT_  32
#define RR_  16
#define PARW 0.6f
#define MAXL 5.0f

// Flat sizes
#define SLOT_N 67108864L   // B*NY*MT*RR*2
#define LEN_N  33554432L   // B*NY*MT*RR
#define WID_N  33554432L
#define CNT_N  65536L      // B*NY

#define AS1 __attribute__((address_space(1)))
#define AS3 __attribute__((address_space(3)))

// Native 16-byte vector types (layout-identical to HIP float4/int4; required
// by the nontemporal builtins and matching the async-LDS builtins' pointee).
typedef int   vint4_t   __attribute__((vector_size(16)));
typedef float vfloat4_t __attribute__((vector_size(16)));

#if defined(__HIP_DEVICE_COMPILE__) && defined(__gfx1250__) && \
    __has_builtin(__builtin_amdgcn_global_load_async_to_lds_b128) && \
    __has_builtin(__builtin_amdgcn_global_store_async_from_lds_b128) && \
    __has_builtin(__builtin_amdgcn_s_wait_asynccnt)
#define USE_ASYNC_LDS 1
#else
#define USE_ASYNC_LDS 0
#endif

// Global flat addresses are value-identical in generic and AS1.
#define GPTR(p) ((AS1 vint4_t*)(unsigned long long)(const void*)(p))
// Generic LDS pointers carry the 32-bit LDS offset in their low bits.
#define LPTR(p) ((AS3 vint4_t*)(unsigned)(unsigned long long)(const void*)(p))

// ---------------------------------------------------------------------------
// int32 -> float32 value-converting streaming copy (128-bit per lane,
// non-temporal: these arrays exceed L2 and are written once).
// ---------------------------------------------------------------------------
__global__ __launch_bounds__(256) void copy_i2f(const vint4_t* __restrict__ src,
                                                vfloat4_t* __restrict__ dst,
                                                long n4) {
    long i = (long)blockIdx.x * blockDim.x + threadIdx.x;
    long stride = (long)gridDim.x * blockDim.x;
    for (; i < n4; i += stride) {
        vint4_t v = __builtin_nontemporal_load(src + i);
        vfloat4_t f = __builtin_convertvector(v, vfloat4_t);
        __builtin_nontemporal_store(f, dst + i);
    }
}

// ---------------------------------------------------------------------------
// f32 -> f32 streaming copy staged through LDS with the CDNA5 async engine.
// 4 tiles per ASYNCcnt wait group: 4 async loads -> s_wait_asynccnt 0 ->
// 4 async stores -> s_wait_asynccnt 0. Fallback: plain non-temporal b128 copy.
// ---------------------------------------------------------------------------
#define ATILES 4
__global__ __launch_bounds__(256) void copy_f32(const vfloat4_t* __restrict__ src,
                                                vfloat4_t* __restrict__ dst,
                                                long n4) {
#if USE_ASYNC_LDS
    __shared__ vfloat4_t buf[256 * ATILES];
    long tid = (long)blockIdx.x * blockDim.x + threadIdx.x;
    long nthreads = (long)gridDim.x * blockDim.x;
    for (long i = tid; i < n4; i += nthreads * ATILES) {
        #pragma unroll
        for (int t = 0; t < ATILES; ++t) {
            long idx = i + (long)t * nthreads;
            if (idx < n4)
                __builtin_amdgcn_global_load_async_to_lds_b128(
                    GPTR(src + idx), LPTR(&buf[threadIdx.x + 256 * t]), 0, 0);
        }
        __builtin_amdgcn_s_wait_asynccnt(0);   // all 4 tiles resident in LDS
        #pragma unroll
        for (int t = 0; t < ATILES; ++t) {
            long idx = i + (long)t * nthreads;
            if (idx < n4)
                __builtin_amdgcn_global_store_async_from_lds_b128(
                    GPTR(dst + idx), LPTR(&buf[threadIdx.x + 256 * t]), 0, 0);
        }
        __builtin_amdgcn_s_wait_asynccnt(0);   // LDS slots reusable
    }
#else
    long i = (long)blockIdx.x * blockDim.x + threadIdx.x;
    long stride = (long)gridDim.x * blockDim.x;
    for (; i < n4; i += stride) {
        vfloat4_t v = __builtin_nontemporal_load(src + i);
        __builtin_nontemporal_store(v, dst + i);
    }
#endif
}

// ---------------------------------------------------------------------------
// Per-batch update: one lane per batch element.
// Overwrites the (already copied) output values; all values are computed
// from the ORIGINAL inputs so there is no output RMW hazard.
// ---------------------------------------------------------------------------
__global__ __launch_bounds__(256) void update_kernel(
        const int*   __restrict__ block_selection,
        const float* __restrict__ block_size,      // (B,1,5)
        const int*   __restrict__ yard_selection,
        const int*   __restrict__ barge_count,     // (B,NY)
        const float* __restrict__ lengths_in,      // (B,NY,MT,RR)
        const float* __restrict__ widths_in,       // (B,NY,MT,RR)
        float* __restrict__ out) {
    int b = blockIdx.x * blockDim.x + threadIdx.x;
    if (b >= B_) return;

    int y = yard_selection[b];
    int s = barge_count[b * NY_ + y];
    long base = (((long)(b * NY_ + y)) * MT_ + s) * RR_;   // 64B-aligned row

    float w[RR_], l[RR_];
    {
        const vfloat4_t* wp = reinterpret_cast<const vfloat4_t*>(widths_in + base);
        const vfloat4_t* lp = reinterpret_cast<const vfloat4_t*>(lengths_in + base);
        #pragma unroll
        for (int q = 0; q < 4; ++q) {
            vfloat4_t a = wp[q]; vfloat4_t c = lp[q];
            #pragma unroll
            for (int e = 0; e < 4; ++e) {
                w[4*q+e] = a[e];
                l[4*q+e] = c[e];
            }
        }
    }

    bool has_par_space = false, has_valid = false;
    float maxv = -1.0f;      // sentinel used by reference (par_lengths filled with -1)
    int   max_idx = 0;
    float total = 0.0f;
    #pragma unroll
    for (int i = 0; i < RR_; ++i) {
        bool isp = (w[i] <= PARW);
        has_par_space |= isp;
        has_valid     |= (isp && (w[i] != 0.0f));
        float pl = isp ? l[i] : -1.0f;
        if (pl > maxv) { maxv = pl; max_idx = i; }   // strict '>' => first max (jnp.argmax)
        total += l[i];
    }

    float blen = block_size[b * 5 + 0];
    float bwid = block_size[b * 5 + 1];
    bool par_alloc    = (bwid <= PARW) && has_par_space;
    bool has_existing = has_valid && par_alloc;
    bool dir_mask     = has_existing && (maxv >= blen);
    bool recons       = has_existing && (maxv <  blen);
    bool allocable    = (total - maxv + blen) <= MAXL;
    bool ra  = recons &&  allocable;
    bool rna = recons && !allocable;
    float sel = (float)block_selection[b];

    float* out_slot = out;
    float* out_len  = out + SLOT_N;
    float* out_wid  = out + SLOT_N + LEN_N;
    float* out_cnt  = out + SLOT_N + LEN_N + WID_N;

    if (dir_mask || ra) {
        long sidx = (base + max_idx) * 2 + 1;          // [b,y,s,max_idx,1]
        out_slot[sidx] = sel;
        out_wid[base + max_idx] = w[max_idx] + bwid;
        if (ra) out_len[base + max_idx] = blen;
    }
    if (rna) {
        out_cnt[b * NY_ + y] = (float)(s + 1);         // barge_count += 1 (no clamp)
        int s2 = (s + 1 < MT_ - 1) ? (s + 1) : (MT_ - 1);
        long base2 = (((long)(b * NY_ + y)) * MT_ + s2) * RR_;
        out_slot[base2 * 2 + 0] = sel;                 // [b,y,s2,0,0]
        out_len[base2] = blen;
        out_wid[base2] = widths_in[base2] + bwid;
    }
}

// ---------------------------------------------------------------------------
extern "C" void kernel_launch(void* const* d_in, const int* in_sizes, int n_in,
                              void* d_out, int out_size, void* d_ws, size_t ws_size,
                              hipStream_t stream) {
    const int*   sel    = (const int*)  d_in[0];   // block_selection (B,)
    const float* bsize  = (const float*)d_in[1];   // block_size (B,1,5)
    const int*   ysel   = (const int*)  d_in[2];   // yard_selection (B,)
    const int*   bcount = (const int*)  d_in[3];   // barge_count (B,NY)
    const int*   bslot  = (const int*)  d_in[4];   // barge_slot (B,NY,MT,RR,2)
    const float* blen   = (const float*)d_in[5];   // block_lengths_on_barge
    const float* bwid   = (const float*)d_in[6];   // block_width_on_barge
    float* out = (float*)d_out;

    // 1) Bulk copies (output regions are disjoint; stream-ordered before update).
    copy_i2f<<<4096, 256, 0, stream>>>((const vint4_t*)bslot,
                                       (vfloat4_t*)out, SLOT_N / 4);
    copy_f32<<<4096, 256, 0, stream>>>((const vfloat4_t*)blen,
                                       (vfloat4_t*)(out + SLOT_N), LEN_N / 4);
    copy_f32<<<4096, 256, 0, stream>>>((const vfloat4_t*)bwid,
                                       (vfloat4_t*)(out + SLOT_N + LEN_N), WID_N / 4);
    copy_i2f<<<64, 256, 0, stream>>>((const vint4_t*)bcount,
                                     (vfloat4_t*)(out + SLOT_N + LEN_N + WID_N), CNT_N / 4);

    // 2) Scattered per-batch updates on top of the copied state.
    update_kernel<<<B_ / 256, 256, 0, stream>>>(sel, bsize, ysel, bcount,
                                                blen, bwid, out);
}